// SlidingWindowAttention_84490596647140
// MI455X (gfx1250) — compile-verified
//
#include <hip/hip_runtime.h>
#include <hip/hip_bf16.h>
#include <math.h>

// ---------------------------------------------------------------------------
// Sliding-window dilated causal attention, fp32, CDNA5 (gfx1250), wave32.
//   D_MODEL=1024, H=16, DH=64, WINDOW=256, DILATION=2, SEQ=2048, BSZ=2
// Pipeline:
//   1) gemm_xwT<0> x3 : q/k/v = x @ W^T + b   -> ws as [b,h,s,dh]
//   2) attn_kernel    : per-parity sliding-window softmax attention
//   3) gemm_xwT<1>    : out = attn @ o_w^T + o_b
// Matrix math: V_WMMA_F32_16X16X4_F32 (f32 to match fp32 reference).
// Staging: GLOBAL_LOAD_ASYNC_TO_LDS_B128 double-buffered (ASYNCcnt),
//          falling back to sync float4 copies if the builtin is absent.
// LDS rows padded to 68 floats: 272B row pitch -> 16B-aligned B128 stores,
// and 68*m mod 64 distinct over 16 lanes -> conflict-free fragment reads.
// ---------------------------------------------------------------------------

typedef float v2f __attribute__((ext_vector_type(2)));
typedef float v8f __attribute__((ext_vector_type(8)));
typedef int   v4i __attribute__((vector_size(16)));   // matches builtin pointee

#define SEQ     2048
#define BSZ     2
#define DM      1024
#define NH      16
#define DH      64
#define NROWS   (SEQ * BSZ)   // 4096
#define NEGINF  (-1.0e30f)
#define LPAD    68            // LDS row pitch in floats

#define HAS_ASYNC_LDS __has_builtin(__builtin_amdgcn_global_load_async_to_lds_b128)

typedef __attribute__((address_space(1))) v4i* gas_p;   // global (AS1) int4*
typedef __attribute__((address_space(3))) v4i* las_p;   // LDS    (AS3) int4*

static __device__ __forceinline__ void async_copy16(const float* g, float* l) {
#if HAS_ASYNC_LDS
  __builtin_amdgcn_global_load_async_to_lds_b128((gas_p)(void*)g, (las_p)(void*)l,
                                                 0, 0);
#else
  *reinterpret_cast<float4*>(l) = *reinterpret_cast<const float4*>(g);
#endif
}

static __device__ __forceinline__ void wait_async_0() {
#if HAS_ASYNC_LDS
# if __has_builtin(__builtin_amdgcn_s_wait_asynccnt)
  __builtin_amdgcn_s_wait_asynccnt(0);
# else
  asm volatile("s_wait_asynccnt 0x0" ::: "memory");
# endif
#endif
}

static __device__ __forceinline__ void wait_async_8() {
#if HAS_ASYNC_LDS
# if __has_builtin(__builtin_amdgcn_s_wait_asynccnt)
  __builtin_amdgcn_s_wait_asynccnt(8);
# else
  asm volatile("s_wait_asynccnt 0x8" ::: "memory");
# endif
#endif
}

static __device__ __forceinline__ v8f wmma_f32(v2f a, v2f b, v8f c) {
  // (neg_a, A, neg_b, B, c_mod, C, reuse_a, reuse_b)
  return __builtin_amdgcn_wmma_f32_16x16x4_f32(
      false, a, false, b, (short)0, c, false, false);
}

// ---------------------------------------------------------------------------
// GEMM: Out[4096,1024] = X[4096,1024] @ W[1024,1024]^T + bias
// Block: 128 threads (4 waves). Block tile: 64 rows x 32 cols.
// Wave tile: 16 x 32 (two accumulators share the A fragment).
// K staged in double-buffered 64-wide LDS chunks via async-LDS copies.
// MODE 0: scatter to [b,h,s,dh] (q/k/v).  MODE 1: row-major.
// ---------------------------------------------------------------------------
static __device__ __forceinline__ void stage_gemm(
    const float* __restrict__ X, const float* __restrict__ W,
    float (*As)[LPAD], float (*Bs)[LPAD],
    int rbase, int cbase, int kb, int tid) {
#pragma unroll
  for (int l = 0; l < 8; ++l) {           // A: 64 rows x 16 float4
    int lin = tid + l * 128;              // 0..1023
    int row = lin >> 4;
    int c4  = (lin & 15) << 2;
    async_copy16(&X[(size_t)(rbase + row) * DM + kb + c4], &As[row][c4]);
  }
#pragma unroll
  for (int l = 0; l < 4; ++l) {           // B: 32 rows x 16 float4
    int lin = tid + l * 128;              // 0..511
    int row = lin >> 4;
    int c4  = (lin & 15) << 2;
    async_copy16(&W[(size_t)(cbase + row) * DM + kb + c4], &Bs[row][c4]);
  }
}

template <int MODE>
__global__ void __launch_bounds__(128)
gemm_xwT(const float* __restrict__ X, const float* __restrict__ W,
         const float* __restrict__ bias, float* __restrict__ Out) {
  __shared__ float As[2][64][LPAD];
  __shared__ float Bs[2][32][LPAD];

  const int tid  = threadIdx.x;
  const int lane = tid & 31;
  const int wid  = tid >> 5;
  const int half = lane >> 4;
  const int m    = lane & 15;

  const int rbase = blockIdx.x * 64;
  const int cbase = blockIdx.y * 32;

  v8f acc0, acc1;
#pragma unroll
  for (int i = 0; i < 8; ++i) { acc0[i] = 0.0f; acc1[i] = 0.0f; }

  stage_gemm(X, W, As[0], Bs[0], rbase, cbase, 0, tid);

  int p = 0;
  for (int kb = 0; kb < DM; kb += 64) {
    wait_async_0();
    __syncthreads();                      // chunk kb ready in buffer p
    if (kb + 64 < DM)
      stage_gemm(X, W, As[p ^ 1], Bs[p ^ 1], rbase, cbase, kb + 64, tid);

#pragma unroll
    for (int kk = 0; kk < 64; kk += 4) {
      const int ko = kk + 2 * half;       // lane-half picks K pair {0,1}/{2,3}
      v2f a  = { As[p][wid * 16 + m][ko], As[p][wid * 16 + m][ko + 1] };
      v2f b0 = { Bs[p][m][ko],            Bs[p][m][ko + 1] };
      v2f b1 = { Bs[p][16 + m][ko],       Bs[p][16 + m][ko + 1] };
      acc0 = wmma_f32(a, b0, acc0);
      acc1 = wmma_f32(a, b1, acc1);
    }
    p ^= 1;
  }

  const int col0 = cbase + m, col1 = cbase + 16 + m;
  const float bc0 = bias[col0], bc1 = bias[col1];
#pragma unroll
  for (int v = 0; v < 8; ++v) {
    const int M   = v + 8 * half;
    const int row = rbase + wid * 16 + M;
    const float v0 = acc0[v] + bc0;
    const float v1 = acc1[v] + bc1;
    if (MODE == 0) {
      const int s = row >> 1, b = row & 1;
      Out[(((size_t)(b * NH + (col0 >> 6))) * SEQ + s) * DH + (col0 & 63)] = v0;
      Out[(((size_t)(b * NH + (col1 >> 6))) * SEQ + s) * DH + (col1 & 63)] = v1;
    } else {
      Out[(size_t)row * DM + col0] = v0;
      Out[(size_t)row * DM + col1] = v1;
    }
  }
}

// ---------------------------------------------------------------------------
// Attention: one wave per (b, h, parity, 16-query tile).
// Dilation=2 => each parity subsequence is a standard causal sliding window
// of width 256.  Keys j: t(j) = s0 - 510 + 2j, j in [0,271]; band j in [i,i+255].
// ---------------------------------------------------------------------------
static __device__ __forceinline__ void stage_kv16(
    const float* __restrict__ base, float (*dst)[LPAD], int s0, int jt, int lane) {
#pragma unroll
  for (int l = 0; l < 8; ++l) {           // 16 keys x 16 float4
    int lin = lane + (l << 5);            // 0..255
    int row = lin >> 4;
    int c4  = (lin & 15) << 2;
    int t   = s0 - 510 + 2 * (jt * 16 + row);
    int tc  = t < 0 ? 0 : (t > SEQ - 1 ? SEQ - 1 : t);   // clamped; masked later
    async_copy16(&base[(size_t)tc * DH + c4], &dst[row][c4]);
  }
}

__global__ void __launch_bounds__(32)
attn_kernel(const float* __restrict__ q, const float* __restrict__ k,
            const float* __restrict__ v, const float* __restrict__ beta,
            float* __restrict__ out) {
  __shared__ float Qs[16][LPAD];
  __shared__ float KV[2][16][LPAD];
  __shared__ float Ps[16][273];           // odd pitch -> conflict-free A frags
  __shared__ float Linv[16];

  const int bid  = blockIdx.x;
  const int tile = bid & 63;
  const int par  = (bid >> 6) & 1;
  const int h    = (bid >> 7) & 15;
  const int b    = bid >> 11;

  const int lane = threadIdx.x;
  const int half = lane >> 4;
  const int m    = lane & 15;

  const int s0 = par + 32 * tile;
  const size_t headoff = ((size_t)(b * NH + h)) * SEQ * DH;
  const float* Q = q + headoff;
  const float* K = k + headoff;
  const float* V = v + headoff;

  const float scale = 0.125f * __expf(-beta[h]);   // 1/(sqrt(DH)*exp(beta))

  // ---- stage Q tile (pure copy; scale applied post-WMMA) ----
#pragma unroll
  for (int l = 0; l < 8; ++l) {
    int lin = lane + (l << 5);
    int row = lin >> 4;
    int c4  = (lin & 15) << 2;
    async_copy16(&Q[(size_t)(s0 + 2 * row) * DH + c4], &Qs[row][c4]);
  }
  stage_kv16(K, KV[0], s0, 0, lane);

  // ---- scores: 17 key tiles, double-buffered K staging ----
  for (int jt = 0; jt < 17; ++jt) {
    if (jt < 16) stage_kv16(K, KV[(jt + 1) & 1], s0, jt + 1, lane);
    if (jt < 16) wait_async_8(); else wait_async_0();

    v8f c;
#pragma unroll
    for (int i = 0; i < 8; ++i) c[i] = 0.0f;
#pragma unroll
    for (int kk = 0; kk < DH; kk += 4) {
      const int ko = kk + 2 * half;
      v2f a  = { Qs[m][ko],            Qs[m][ko + 1] };
      v2f bb = { KV[jt & 1][m][ko],    KV[jt & 1][m][ko + 1] };  // B[d][n]=K[n][d]
      c = wmma_f32(a, bb, c);
    }

#pragma unroll
    for (int vv = 0; vv < 8; ++vv) {
      const int i = vv + 8 * half;
      const int j = jt * 16 + m;
      const int t = s0 - 510 + 2 * j;
      const bool ok = (t >= 0) && (j >= i) && (j <= i + 255);
      Ps[i][j] = ok ? c[vv] * scale : NEGINF;
    }
  }

  // ---- per-row softmax over j = 0..271 (masked -> exp = 0) ----
  if (lane < 16) {
    const int i = lane;
    float mx = NEGINF;
    for (int j = 0; j < 272; ++j) mx = fmaxf(mx, Ps[i][j]);
    float sum = 0.0f;
    for (int j = 0; j < 272; ++j) {
      float e = __expf(Ps[i][j] - mx);
      Ps[i][j] = e;
      sum += e;
    }
    Linv[i] = 1.0f / sum;
  }
  __syncthreads();   // single-wave: lowers to fence/nop, reconverges EXEC

  // ---- O = P x V : 4 column tiles, double-buffered V staging ----
  v8f acc[4];
#pragma unroll
  for (int nt = 0; nt < 4; ++nt)
#pragma unroll
    for (int i = 0; i < 8; ++i) acc[nt][i] = 0.0f;

  stage_kv16(V, KV[0], s0, 0, lane);
  for (int jt = 0; jt < 17; ++jt) {
    if (jt < 16) stage_kv16(V, KV[(jt + 1) & 1], s0, jt + 1, lane);
    if (jt < 16) wait_async_8(); else wait_async_0();

#pragma unroll
    for (int nt = 0; nt < 4; ++nt) {
#pragma unroll
      for (int k4 = 0; k4 < 4; ++k4) {
        const int jb = jt * 16 + k4 * 4 + 2 * half;   // global j for A frag
        const int kl = k4 * 4 + 2 * half;             // local key row for B frag
        v2f a  = { Ps[m][jb], Ps[m][jb + 1] };
        v2f bb = { KV[jt & 1][kl][nt * 16 + m], KV[jt & 1][kl + 1][nt * 16 + m] };
        acc[nt] = wmma_f32(a, bb, acc[nt]);
      }
    }
  }

  // ---- normalize, write to [s,b,D] so the output projection is a plain GEMM --
#pragma unroll
  for (int nt = 0; nt < 4; ++nt) {
#pragma unroll
    for (int vv = 0; vv < 8; ++vv) {
      const int i = vv + 8 * half;
      const int s = s0 + 2 * i;
      const int col = h * DH + nt * 16 + m;
      out[((size_t)s * BSZ + b) * DM + col] = acc[nt][vv] * Linv[i];
    }
  }
}

// ---------------------------------------------------------------------------
extern "C" void kernel_launch(void* const* d_in, const int* in_sizes, int n_in,
                              void* d_out, int out_size, void* d_ws, size_t ws_size,
                              hipStream_t stream) {
  const float* x    = (const float*)d_in[0];
  const float* q_w  = (const float*)d_in[1];
  const float* q_b  = (const float*)d_in[2];
  const float* k_w  = (const float*)d_in[3];
  const float* k_b  = (const float*)d_in[4];
  const float* v_w  = (const float*)d_in[5];
  const float* v_b  = (const float*)d_in[6];
  const float* o_w  = (const float*)d_in[7];
  const float* o_b  = (const float*)d_in[8];
  const float* beta = (const float*)d_in[9];

  const size_t CHUNK = (size_t)BSZ * NH * SEQ * DH;  // 4,194,304 floats
  float* qbuf = (float*)d_ws;
  float* kbuf = qbuf + CHUNK;
  float* vbuf = kbuf + CHUNK;
  float* abuf = vbuf + CHUNK;

  dim3 gblk(128);
  dim3 ggrid(NROWS / 64, DM / 32);   // 64 x 32

  gemm_xwT<0><<<ggrid, gblk, 0, stream>>>(x, q_w, q_b, qbuf);
  gemm_xwT<0><<<ggrid, gblk, 0, stream>>>(x, k_w, k_b, kbuf);
  gemm_xwT<0><<<ggrid, gblk, 0, stream>>>(x, v_w, v_b, vbuf);

  attn_kernel<<<dim3(4096), dim3(32), 0, stream>>>(qbuf, kbuf, vbuf, beta, abuf);

  gemm_xwT<1><<<ggrid, gblk, 0, stream>>>(abuf, o_w, o_b, (float*)d_out);
}